// gat_layer_18949395710229
// MI455X (gfx1250) — compile-verified
//
#include <hip/hip_runtime.h>
#include <math.h>

// GAT layer for MI455X (gfx1250, wave32).
// Pipeline: [init] -> [WMMA f32 GEMM h=X@W] -> [node dots hs,hd] ->
//           [edge segment-max] -> [decode max] -> [edge exp-sum] ->
//           [edge scatter out[src]+=attn*h[dst]] -> [ELU].
// Scatter is the bottleneck (~400MB of L2-resident read+atomic traffic);
// the GEMM (0.82 GFLOP) runs exact f32 via V_WMMA_F32_16X16X4_F32.

typedef __attribute__((ext_vector_type(2))) float v2f;
typedef __attribute__((ext_vector_type(8))) float v8f;

#define NEG_SLOPE 0.2f

// Monotonic float->int key for atomicMax-based float max.
__device__ __forceinline__ int fkey(float f) {
  int i = __float_as_int(f);
  return i >= 0 ? i : (i ^ 0x7FFFFFFF);
}
__device__ __forceinline__ float funkey(int k) {
  return __int_as_float(k >= 0 ? k : (k ^ 0x7FFFFFFF));
}
#define NEG_INF_KEY 0x807FFFFF  // fkey(-inf)

// ---------------- K0: zero out + init per-node accumulators ----------------
__global__ void k_init(float* __restrict__ out, int* __restrict__ m_enc,
                       float* __restrict__ ssum, int N, int total) {
  int i = blockIdx.x * blockDim.x + threadIdx.x;
  if (i < total) out[i] = 0.0f;
  if (i < N) { m_enc[i] = NEG_INF_KEY; ssum[i] = 0.0f; }
}

// ---------------- K1: h = X @ W via V_WMMA_F32_16X16X4_F32 -----------------
// One wave per 16-row tile; 4 accumulators cover N=64. W is staged transposed
// in LDS (32 KB of the 320 KB WGP pool) so B fragments are single b64 reads.
// N (=50000) is a multiple of 16, so tile guards are wave-uniform and EXEC
// stays all-ones across every WMMA (ISA requirement).
__global__ void __launch_bounds__(256) k_gemm(const float* __restrict__ X,
                                              const float* __restrict__ W,
                                              float* __restrict__ H, int N) {
  __shared__ float s_wt[64 * 128];  // s_wt[n*128 + k] = W[k*64 + n]
  const int tid = threadIdx.x;
  for (int i = tid; i < 64 * 128; i += 256) {
    int k = i & 127, n = i >> 7;
    s_wt[n * 128 + k] = W[k * 64 + n];
  }
  __syncthreads();

  const int wave = tid >> 5;
  const int lane = tid & 31;
  const int half = lane >> 4;   // K-half select (A: K 0,1 vs 2,3)
  const int l    = lane & 15;   // M for A, N for B/C/D
  const int tile = blockIdx.x * 8 + wave;
  const int row0 = tile * 16;
  if (row0 >= N) return;        // wave-uniform

  v8f acc0 = {}, acc1 = {}, acc2 = {}, acc3 = {};
  const float* xrow = X + (long)(row0 + l) * 128 + 2 * half;
  const float* wt0 = s_wt + (0 * 16 + l) * 128 + 2 * half;
  const float* wt1 = s_wt + (1 * 16 + l) * 128 + 2 * half;
  const float* wt2 = s_wt + (2 * 16 + l) * 128 + 2 * half;
  const float* wt3 = s_wt + (3 * 16 + l) * 128 + 2 * half;

#pragma unroll
  for (int k = 0; k < 128; k += 4) {
    v2f a  = *(const v2f*)(xrow + k);
    v2f b0 = *(const v2f*)(wt0 + k);
    v2f b1 = *(const v2f*)(wt1 + k);
    v2f b2 = *(const v2f*)(wt2 + k);
    v2f b3 = *(const v2f*)(wt3 + k);
    acc0 = __builtin_amdgcn_wmma_f32_16x16x4_f32(false, a, false, b0, (short)0, acc0, false, false);
    acc1 = __builtin_amdgcn_wmma_f32_16x16x4_f32(false, a, false, b1, (short)0, acc1, false, false);
    acc2 = __builtin_amdgcn_wmma_f32_16x16x4_f32(false, a, false, b2, (short)0, acc2, false, false);
    acc3 = __builtin_amdgcn_wmma_f32_16x16x4_f32(false, a, false, b3, (short)0, acc3, false, false);
  }

  // C/D layout: VGPR i holds row M=i (lanes 0-15) and M=i+8 (lanes 16-31).
#pragma unroll
  for (int i = 0; i < 8; ++i) {
    int m = half * 8 + i;
    float* hrow = H + (long)(row0 + m) * 64 + l;
    hrow[0]  = acc0[i];
    hrow[16] = acc1[i];
    hrow[32] = acc2[i];
    hrow[48] = acc3[i];
  }
}

// ---------------- K2: hs[i] = h[i]·a_src, hd[i] = h[i]·a_dst ---------------
__global__ void k_scores(const float* __restrict__ H, const float* __restrict__ a,
                         float* __restrict__ hs, float* __restrict__ hd, int N) {
  int i = blockIdx.x * blockDim.x + threadIdx.x;
  if (i >= N) return;
  const float* hrow = H + (long)i * 64;
  float s = 0.0f, d = 0.0f;
#pragma unroll
  for (int j = 0; j < 64; ++j) {
    float v = hrow[j];
    s = fmaf(v, a[j], s);
    d = fmaf(v, a[64 + j], d);
  }
  hs[i] = s;
  hd[i] = d;
}

__device__ __forceinline__ float edge_score(const float* hs, const float* hd,
                                            int s, int d) {
  float sc = hs[s] + hd[d];
  return sc > 0.0f ? sc : NEG_SLOPE * sc;
}

// ---------------- K3: segment max over src ---------------------------------
__global__ void k_edge_max(const int* __restrict__ edge, const float* __restrict__ hs,
                           const float* __restrict__ hd, int* __restrict__ m_enc, int E) {
  int e = blockIdx.x * blockDim.x + threadIdx.x;
  if (e >= E) return;
  int s = edge[e], d = edge[E + e];
  atomicMax(&m_enc[s], fkey(edge_score(hs, hd, s, d)));
}

// ---------------- K4: decode max in place; non-finite -> 0 -----------------
__global__ void k_finalize_max(int* __restrict__ m_enc, int N) {
  int i = blockIdx.x * blockDim.x + threadIdx.x;
  if (i >= N) return;
  float f = funkey(m_enc[i]);
  bool finite = (__float_as_int(f) & 0x7F800000) != 0x7F800000;
  ((float*)m_enc)[i] = finite ? f : 0.0f;
}

// ---------------- K5: segment sum of exp(e - m[src]) -----------------------
__global__ void k_edge_sum(const int* __restrict__ edge, const float* __restrict__ hs,
                           const float* __restrict__ hd, const float* __restrict__ m,
                           float* __restrict__ ssum, int E) {
  int e = blockIdx.x * blockDim.x + threadIdx.x;
  if (e >= E) return;
  int s = edge[e], d = edge[E + e];
  float ex = expf(edge_score(hs, hd, s, d) - m[s]);
  atomicAdd(&ssum[s], ex);
}

// ---------------- K6: out[src] += attn * h[dst] ----------------------------
// One wave per edge, 2 features per lane: coalesced b64 reads of h[dst] and
// lane-parallel f32 atomics into out[src] (L2-resident: 12.8 MB << 192 MB).
__global__ void __launch_bounds__(256) k_edge_scatter(
    const int* __restrict__ edge, const float* __restrict__ hs,
    const float* __restrict__ hd, const float* __restrict__ m,
    const float* __restrict__ ssum, const float* __restrict__ H,
    float* __restrict__ out, int E) {
  int e = blockIdx.x * 8 + (threadIdx.x >> 5);
  if (e >= E) return;
  int lane = threadIdx.x & 31;
  int s = edge[e], d = edge[E + e];
  float attn = expf(edge_score(hs, hd, s, d) - m[s]) / ssum[s];
  const float* hrow = H + (long)d * 64 + lane * 2;
  float* orow = out + (long)s * 64 + lane * 2;
  v2f hv = *(const v2f*)hrow;
  atomicAdd(&orow[0], attn * hv.x);
  atomicAdd(&orow[1], attn * hv.y);
}

// ---------------- K7: ELU in place -----------------------------------------
__global__ void k_elu(float* __restrict__ out, int total) {
  int i = blockIdx.x * blockDim.x + threadIdx.x;
  if (i >= total) return;
  float v = out[i];
  out[i] = v > 0.0f ? v : expm1f(v);
}

extern "C" void kernel_launch(void* const* d_in, const int* in_sizes, int n_in,
                              void* d_out, int out_size, void* d_ws, size_t ws_size,
                              hipStream_t stream) {
  const float* X    = (const float*)d_in[0];   // (N, 128)
  const int*   edge = (const int*)d_in[1];     // (2, E) int32 (jax x64 disabled)
  const float* W    = (const float*)d_in[2];   // (128, 64)
  const float* a    = (const float*)d_in[3];   // (128, 1)
  float* out = (float*)d_out;                  // (N, 64)

  const int N = in_sizes[0] / 128;
  const int E = in_sizes[1] / 2;
  const int total = N * 64;

  // Workspace layout: H (N*64) | hs (N) | hd (N) | m_enc (N, int->float) | ssum (N)
  float* H    = (float*)d_ws;
  float* hs   = H + (size_t)N * 64;
  float* hd   = hs + N;
  int*   m_enc = (int*)(hd + N);
  float* ssum = (float*)(m_enc + N);

  k_init<<<(total + 255) / 256, 256, 0, stream>>>(out, m_enc, ssum, N, total);

  const int tiles = (N + 15) / 16;              // 16-row WMMA tiles (N%16==0)
  k_gemm<<<(tiles + 7) / 8, 256, 0, stream>>>(X, W, H, N);

  k_scores<<<(N + 255) / 256, 256, 0, stream>>>(H, a, hs, hd, N);
  k_edge_max<<<(E + 255) / 256, 256, 0, stream>>>(edge, hs, hd, m_enc, E);
  k_finalize_max<<<(N + 255) / 256, 256, 0, stream>>>(m_enc, N);
  k_edge_sum<<<(E + 255) / 256, 256, 0, stream>>>(edge, hs, hd, (const float*)m_enc, ssum, E);
  k_edge_scatter<<<(E + 7) / 8, 256, 0, stream>>>(edge, hs, hd, (const float*)m_enc,
                                                  ssum, H, out, E);
  k_elu<<<(total + 255) / 256, 256, 0, stream>>>(out, total);
}